// ClustAttn_75548474737129
// MI455X (gfx1250) — compile-verified
//
#include <hip/hip_runtime.h>
#include <math.h>

typedef __attribute__((ext_vector_type(2))) float v2f;
typedef __attribute__((ext_vector_type(8))) float v8f;

#define BSZ   4
#define SPAT  10000
#define DM    256
#define NTOT  (BSZ * SPAT)   // 40000

// ---------------- ws layout (floats) ----------------
// [0      , 40000) : logits            (later reused for step_raw)
// [40000  , 80000) : ent0 (raw entropy)
// [80000  ,120000) : ents (gauss-smoothed entropy, kept for pooling)
// [120000 ,120004) : bmax[4]
// [120004 ,120008) : bsum[4]
// ints at float-offset 120008: nruns[4]
// ints at float-offset 120012: params[4] = {clst, lo, nc, pad}
// [120016 , ...  ) : cmax[4*nc], csum[4*nc]
#define OFF_LOGITS 0
#define OFF_ENT0   40000
#define OFF_ENTS   80000
#define OFF_RAW    0          // reuse logits slot (logits dead after entropy)
#define OFF_BMAX   120000
#define OFF_BSUM   120004
#define OFF_NRUNS  120008
#define OFF_PARAMS 120012
#define OFF_CLUST  120016

// ---------- 1) logits = k_row · W + b  via V_WMMA_F32_16X16X4_F32 ----------
// Wave handles 16 rows (M), K=256 in chunks of 4, only N=0 column of B is live.
__global__ void ca_logits_wmma(const float* __restrict__ K,
                               const float* __restrict__ Ww,
                               const float* __restrict__ Wb,
                               float* __restrict__ logits, int nrows) {
  int wid  = (blockIdx.x * blockDim.x + threadIdx.x) >> 5;
  int lane = threadIdx.x & 31;
  int s0 = wid * 16;
  if (s0 >= nrows) return;                 // wave-uniform
  int m  = lane & 15;                      // A row for this lane
  int kk = (lane >> 4) << 1;               // K sub-offset: 0 (lo half) / 2 (hi half)
  const float* arow = K + (size_t)(s0 + m) * DM;
  float bias = Wb[0];
  v8f c = {0.f,0.f,0.f,0.f,0.f,0.f,0.f,0.f};
  for (int k0 = 0; k0 < DM; k0 += 4) {
    v2f a, b;
    a.x = arow[k0 + kk];                   // A[m][k0+kk]
    a.y = arow[k0 + kk + 1];               // A[m][k0+kk+1]
    float w0 = Ww[k0 + kk];
    float w1 = Ww[k0 + kk + 1];
    b.x = (m == 0) ? w0 : 0.0f;            // B[k0+kk][0]   (only column 0 live)
    b.y = (m == 0) ? w1 : 0.0f;            // B[k0+kk+1][0]
    c = __builtin_amdgcn_wmma_f32_16x16x4_f32(false, a, false, b, (short)0, c,
                                              false, false);
  }
  // D[r][0]: lane0 -> rows s0+r, lane16 -> rows s0+8+r
  if (m == 0) {
    int base = s0 + ((lane >> 4) << 3);
#pragma unroll
    for (int r = 0; r < 8; ++r) logits[base + r] = c[r] + bias;
  }
}

// ---------- 2) per-batch softmax stats (max, sum-exp) ----------
__global__ void ca_softmax_stats(const float* __restrict__ logits,
                                 float* __restrict__ bmax,
                                 float* __restrict__ bsum) {
  int b = blockIdx.x;
  const float* x = logits + b * SPAT;
  __shared__ float red[1024];
  int t = threadIdx.x;
  float mx = -3.4e38f;
  for (int i = t; i < SPAT; i += 1024) mx = fmaxf(mx, x[i]);
  red[t] = mx; __syncthreads();
  for (int s = 512; s > 0; s >>= 1) {
    if (t < s) red[t] = fmaxf(red[t], red[t + s]);
    __syncthreads();
  }
  mx = red[0]; __syncthreads();
  float sm = 0.f;
  for (int i = t; i < SPAT; i += 1024) sm += expf(x[i] - mx);
  red[t] = sm; __syncthreads();
  for (int s = 512; s > 0; s >>= 1) {
    if (t < s) red[t] += red[t + s];
    __syncthreads();
  }
  if (t == 0) { bmax[b] = mx; bsum[b] = red[0]; }
}

// ---------- 3) entropy in bits ----------
__global__ void ca_entropy(const float* __restrict__ logits,
                           const float* __restrict__ bmax,
                           const float* __restrict__ bsum,
                           float* __restrict__ ent0) {
  int i = blockIdx.x * blockDim.x + threadIdx.x;
  if (i >= NTOT) return;
  int b = i / SPAT;
  float p = expf(logits[i] - bmax[b]) / bsum[b] + 1e-8f;
  ent0[i] = -p * logf(p) * 1.44269504088896340736f;  // /ln(2)
}

// ---------- 4) gaussian SAME conv, K=17, zero-pad per batch row ----------
__global__ void ca_conv_gauss(const float* __restrict__ ent0,
                              const float* __restrict__ g,
                              float* __restrict__ ents) {
  int i = blockIdx.x * blockDim.x + threadIdx.x;
  if (i >= NTOT) return;
  int b = i / SPAT, s = i - b * SPAT;
  const float* x = ent0 + b * SPAT;
  float acc = 0.f;
#pragma unroll
  for (int k = 0; k < 17; ++k) {
    int j = s + k - 8;
    float v = (j >= 0 && j < SPAT) ? x[j] : 0.0f;
    acc += g[k] * v;
  }
  ents[i] = acc;
}

// ---------- 5) sobel SAME conv, K=3 ----------
__global__ void ca_conv_sobel(const float* __restrict__ ents,
                              const float* __restrict__ sb,
                              float* __restrict__ raw) {
  int i = blockIdx.x * blockDim.x + threadIdx.x;
  if (i >= NTOT) return;
  int b = i / SPAT, s = i - b * SPAT;
  const float* x = ents + b * SPAT;
  float xm = (s >= 1) ? x[s - 1] : 0.f;
  float xp = (s + 1 < SPAT) ? x[s + 1] : 0.f;
  raw[i] = sb[0] * xm + sb[1] * x[s] + sb[2] * xp;
}

// ---------- 6) run count per batch: 1 + #(sign transitions of raw>0) ----------
__global__ void ca_runs(const float* __restrict__ raw, int* __restrict__ nruns) {
  int b = blockIdx.x;
  const float* x = raw + b * SPAT;
  __shared__ int red[256];
  int t = threadIdx.x;
  int cnt = 0;
  for (int i = t + 1; i < SPAT; i += 256)
    cnt += ((x[i] > 0.f) != (x[i - 1] > 0.f)) ? 1 : 0;
  red[t] = cnt; __syncthreads();
  for (int s = 128; s > 0; s >>= 1) {
    if (t < s) red[t] += red[t + s];
    __syncthreads();
  }
  if (t == 0) nruns[b] = red[0] + 1;
}

// ---------- 7) cluster params: clst = max(1, rint(mean_b(SPAT/nruns_b))) ----------
__global__ void ca_params(const int* __restrict__ nruns, int* __restrict__ params) {
  double acc = 0.0;
  for (int b = 0; b < BSZ; ++b) acc += (double)SPAT / (double)nruns[b];
  double mean = acc / (double)BSZ;
  int clst = (int)rint(mean);              // RNE == Python round-half-even
  if (clst < 1) clst = 1;
  int rem = SPAT % clst;
  int lo  = rem >> 1;
  int nc  = (SPAT - rem) / clst;
  params[0] = clst; params[1] = lo; params[2] = nc; params[3] = 0;
}

// ---------- 8) per-cluster softmax stats over smoothed entropy ----------
__global__ void ca_cstats(const float* __restrict__ ents,
                          const int* __restrict__ params,
                          float* __restrict__ cmax, float* __restrict__ csum) {
  int idx = blockIdx.x * blockDim.x + threadIdx.x;
  int nc = params[2];
  if (idx >= BSZ * nc) return;
  int b = idx / nc, c = idx - b * nc;
  int clst = params[0], lo = params[1];
  const float* e = ents + b * SPAT + lo + c * clst;
  float mx = -3.4e38f;
  for (int j = 0; j < clst; ++j) mx = fmaxf(mx, e[j]);
  float sm = 0.f;
  for (int j = 0; j < clst; ++j) sm += expf(e[j] - mx);
  cmax[idx] = mx; csum[idx] = sm;
}

// ---------- 9) weighted pooling via V_WMMA_F32_16X16X4_F32 ----------
// Wave = one (b, cluster, 16-feature tile). A row 0 = weights (lanes 0/16),
// B = 4 segment rows x 16 features. Result in D row 0 (VGPR0, lanes 0-15).
__global__ void ca_pool_wmma(const float* __restrict__ KR,
                             const float* __restrict__ KC,
                             const float* __restrict__ V,
                             const float* __restrict__ ents,
                             const int* __restrict__ params,
                             const float* __restrict__ cmax,
                             const float* __restrict__ csum,
                             float* __restrict__ out) {
  int wid  = (blockIdx.x * blockDim.x + threadIdx.x) >> 5;
  int lane = threadIdx.x & 31;
  int nc = params[2];
  int ft = wid & 15;
  int bc = wid >> 4;
  if (bc >= BSZ * nc) return;              // wave-uniform
  int b = bc / nc, cl = bc - b * nc;
  int clst = params[0], lo = params[1];
  int f0 = ft << 4;
  int m  = lane & 15;                      // B column (feature) / A row
  int hk = (lane >> 4) << 1;               // K sub-offset 0 / 2
  long sbase = (long)b * SPAT + lo + (long)cl * clst;
  const float* eseg = ents + sbase;
  float mx  = cmax[bc];
  float ism = 1.0f / csum[bc];
  v8f cr = {0.f,0.f,0.f,0.f,0.f,0.f,0.f,0.f};
  v8f cc = cr, cv = cr;
  for (int j = 0; j < clst; j += 4) {
    int j0 = j + hk, j1 = j + hk + 1;
    int je0 = (j0 < clst) ? j0 : 0;        // clamp tail (weight forced to 0)
    int je1 = (j1 < clst) ? j1 : 0;
    float e0 = eseg[je0], e1 = eseg[je1];
    v2f a;
    a.x = (m == 0 && j0 < clst) ? expf(e0 - mx) * ism : 0.0f;  // A[0][j0]
    a.y = (m == 0 && j1 < clst) ? expf(e1 - mx) * ism : 0.0f;  // A[0][j1]
    long r0 = (sbase + je0) * DM + f0 + m;
    long r1 = (sbase + je1) * DM + f0 + m;
    v2f bkr, bkc, bv;
    bkr.x = KR[r0]; bkr.y = KR[r1];
    bkc.x = KC[r0]; bkc.y = KC[r1];
    bv.x  = V[r0];  bv.y  = V[r1];
    cr = __builtin_amdgcn_wmma_f32_16x16x4_f32(false, a, false, bkr, (short)0, cr, false, false);
    cc = __builtin_amdgcn_wmma_f32_16x16x4_f32(false, a, false, bkc, (short)0, cc, false, false);
    cv = __builtin_amdgcn_wmma_f32_16x16x4_f32(false, a, false, bv,  (short)0, cv, false, false);
  }
  if (lane < 16) {                         // D[0][n] lives in VGPR0 lanes 0-15
    long tstride = (long)BSZ * nc * DM;
    long obase = ((long)b * nc + cl) * DM + f0 + lane;
    out[obase]               = cr[0];
    out[obase + tstride]     = cc[0];
    out[obase + 2 * tstride] = cv[0];
  }
}

extern "C" void kernel_launch(void* const* d_in, const int* in_sizes, int n_in,
                              void* d_out, int out_size, void* d_ws, size_t ws_size,
                              hipStream_t stream) {
  const float* k_row = (const float*)d_in[0];
  const float* k_col = (const float*)d_in[1];
  const float* value = (const float*)d_in[2];
  const float* W_w   = (const float*)d_in[3];
  const float* W_b   = (const float*)d_in[4];
  const float* gauss = (const float*)d_in[5];
  const float* sobel = (const float*)d_in[6];
  float* out = (float*)d_out;

  float* ws     = (float*)d_ws;
  float* logits = ws + OFF_LOGITS;
  float* ent0   = ws + OFF_ENT0;
  float* ents   = ws + OFF_ENTS;
  float* raw    = ws + OFF_RAW;            // reuses logits slot
  float* bmax   = ws + OFF_BMAX;
  float* bsum   = ws + OFF_BSUM;
  int*   nruns  = (int*)(ws + OFF_NRUNS);
  int*   params = (int*)(ws + OFF_PARAMS);
  int nc_host = out_size / (3 * BSZ * DM); // output shape fixes nc
  float* cmax   = ws + OFF_CLUST;
  float* csum   = cmax + (size_t)BSZ * nc_host;

  // 1) logits (WMMA GEMV)
  {
    int waves = NTOT / 16;                          // 2500
    int blocks = (waves * 32 + 255) / 256;
    ca_logits_wmma<<<blocks, 256, 0, stream>>>(k_row, W_w, W_b, logits, NTOT);
  }
  // 2) batch softmax stats
  ca_softmax_stats<<<BSZ, 1024, 0, stream>>>(logits, bmax, bsum);
  // 3) entropy
  ca_entropy<<<(NTOT + 255) / 256, 256, 0, stream>>>(logits, bmax, bsum, ent0);
  // 4) gaussian conv
  ca_conv_gauss<<<(NTOT + 255) / 256, 256, 0, stream>>>(ent0, gauss, ents);
  // 5) sobel conv (raw overwrites logits slot — logits dead now)
  ca_conv_sobel<<<(NTOT + 255) / 256, 256, 0, stream>>>(ents, sobel, raw);
  // 6) run counts
  ca_runs<<<BSZ, 256, 0, stream>>>(raw, nruns);
  // 7) cluster params
  ca_params<<<1, 1, 0, stream>>>(nruns, params);
  // 8) per-cluster softmax stats
  {
    int n = BSZ * nc_host;
    ca_cstats<<<(n + 255) / 256, 256, 0, stream>>>(ents, params, cmax, csum);
  }
  // 9) pooling (WMMA), one wave per (b, cluster, 16-feature tile)
  {
    long waves = (long)BSZ * nc_host * 16;
    long blocks = (waves * 32 + 255) / 256;
    ca_pool_wmma<<<(int)blocks, 256, 0, stream>>>(k_row, k_col, value, ents,
                                                  params, cmax, csum, out);
  }
}